// GPSModel_26946624815352
// MI455X (gfx1250) — compile-verified
//
#include <hip/hip_runtime.h>
#include <hip/hip_bf16.h>
#include <math.h>

typedef __attribute__((ext_vector_type(2))) float v2f;
typedef __attribute__((ext_vector_type(8))) float v8f;

static constexpr int G_   = 2048;
static constexpr int NPG_ = 24;
static constexpr int EPG_ = 288;
static constexpr int N_   = G_ * NPG_;   // 49152 nodes
static constexpr int E_   = G_ * EPG_;   // 589824 edges
static constexpr int K_   = 32;          // gaussian basis count
static constexpr int EMB_ = 32;
static constexpr int DH_  = 64;
static constexpr int TPW_ = 4;           // edge tiles per wave

// exp(-0.5*a^2) = exp2(q*d^2), q = -0.5*log2(e)/std^2, d = x - mean
static constexpr float NEG_HALF_LOG2E = -0.7213475204444817f;
static constexpr float INV_SQRT_2PI   = 0.3989422804014327f;

// f32 WMMA: D(16x16) = A(16x4) x B(4x16) + C. 8 steps cover K=32.
static __device__ inline v8f wmma_f32(v2f a, v2f b, v8f c) {
    return __builtin_amdgcn_wmma_f32_16x16x4_f32(
        /*neg_a=*/false, a, /*neg_b=*/false, b,
        /*c_mod=*/(short)0, c, /*reuse_a=*/false, /*reuse_b=*/false);
}

// ---------------------------------------------------------------------------
// Node kernel: one block per graph.
//   ssum[i][k] = sum_j exp2(q_k*(x[i][j]-mean_k)^2) * invnorm_k
//   gnf  = atom_emb[z] + ssum @ W_ep + b_ep          (24x32, padded to 32x32)
//   out  = gnf @ W_nodes + b_nodes                   (24x64)
// ---------------------------------------------------------------------------
__global__ __launch_bounds__(256) void node_kernel(
    const int* __restrict__ z, const float* __restrict__ pos,
    const float* __restrict__ atom_emb,
    const float* __restrict__ W_ep, const float* __restrict__ b_ep,
    const float* __restrict__ means, const float* __restrict__ stds,
    const float* __restrict__ gmul, const float* __restrict__ gbias,
    const float* __restrict__ W_n, const float* __restrict__ b_n,
    float* __restrict__ out_node)
{
    __shared__ float sPos[NPG_ * 3];
    __shared__ int   sZ[NPG_];
    __shared__ float sX[NPG_ * NPG_];
    __shared__ float sMean[K_], sQ[K_], sInvNorm[K_], sBep[EMB_];
    __shared__ float sSum[32 * 32];   // padded (rows >=24 are zero)
    __shared__ float sGnf[32 * 32];
    __shared__ float sWep[32 * 32];
    __shared__ float sWn[32 * 64];
    __shared__ float sBn[64];
    __shared__ float sEmb[10 * 32];

    const int g   = blockIdx.x;
    const int tid = threadIdx.x;

    if (tid < NPG_) sZ[tid] = z[g * NPG_ + tid];
    for (int p = tid; p < NPG_ * 3; p += 256) sPos[p] = pos[g * NPG_ * 3 + p];
    if (tid < K_) {
        float s  = fabsf(stds[tid]) + 1e-5f;
        float is = 1.0f / s;
        sMean[tid]    = means[tid];
        sQ[tid]       = NEG_HALF_LOG2E * is * is;
        sInvNorm[tid] = INV_SQRT_2PI * is;
        sBep[tid]     = b_ep[tid];
    }
    for (int p = tid; p < 32 * 32; p += 256) sWep[p] = W_ep[p];
    for (int p = tid; p < 32 * 64; p += 256) sWn[p]  = W_n[p];
    if (tid < 64) sBn[tid] = b_n[tid];
    for (int p = tid; p < 10 * 32; p += 256) sEmb[p] = atom_emb[p];
    for (int p = tid; p < 32 * 32; p += 256) sSum[p] = 0.0f;
    __syncthreads();

    // phase 0: per-pair affine-of-distance table x[i][j]
    for (int p = tid; p < NPG_ * NPG_; p += 256) {
        int i = p / NPG_, j = p % NPG_;
        float dx = sPos[j * 3 + 0] - sPos[i * 3 + 0];
        float dy = sPos[j * 3 + 1] - sPos[i * 3 + 1];
        float dz = sPos[j * 3 + 2] - sPos[i * 3 + 2];
        float sq = dx * dx + dy * dy + dz * dz;
        float dist = (sq == 0.0f) ? 0.0f : sqrtf(sq);
        int et = sZ[i] * 10 + sZ[j];
        sX[p] = gmul[et] * dist + gbias[et];
    }
    __syncthreads();

    // phase 1: ssum[i][k] (768 work items, 24 native exp2 each)
    for (int idx = tid; idx < NPG_ * K_; idx += 256) {
        int i = idx >> 5, k = idx & 31;
        float mn = sMean[k], q = sQ[k];
        float s = 0.0f;
        #pragma unroll
        for (int j = 0; j < NPG_; ++j) {
            float d = sX[i * NPG_ + j] - mn;
            s += exp2f(q * d * d);            // v_exp_f32 (TRANS, co-issues)
        }
        sSum[i * 32 + k] = s * sInvNorm[k];
    }
    __syncthreads();

    const int wave = tid >> 5, lane = tid & 31;
    const int h = lane >> 4, nn = lane & 15;
    const int m0 = wave * 16;   // wave 0 -> rows 0..15, wave 1 -> rows 16..31

    // phase 2: gnf = ssum @ W_ep   (N = 32 -> 2 tiles), + emb + b_ep into LDS
    if (wave < 2) {
        v8f acc0, acc1;
        #pragma unroll
        for (int r = 0; r < 8; ++r) { acc0[r] = 0.0f; acc1[r] = 0.0f; }
        #pragma unroll
        for (int kk = 0; kk < 8; ++kk) {
            int k0 = kk * 4 + 2 * h;
            v2f A; A.x = sSum[(m0 + nn) * 32 + k0];
                   A.y = sSum[(m0 + nn) * 32 + k0 + 1];
            v2f B0; B0.x = sWep[k0 * 32 + nn];       B0.y = sWep[(k0 + 1) * 32 + nn];
            v2f B1; B1.x = sWep[k0 * 32 + 16 + nn];  B1.y = sWep[(k0 + 1) * 32 + 16 + nn];
            acc0 = wmma_f32(A, B0, acc0);
            acc1 = wmma_f32(A, B1, acc1);
        }
        #pragma unroll
        for (int r = 0; r < 8; ++r) {
            int row = m0 + 8 * h + r;                 // C layout: M = r + 8h
            float e0 = 0.0f, e1 = 0.0f;
            if (row < NPG_) {
                int zz = sZ[row];
                e0 = sEmb[zz * 32 + nn];
                e1 = sEmb[zz * 32 + 16 + nn];
            }
            sGnf[row * 32 + nn]      = acc0[r] + sBep[nn] + e0;
            sGnf[row * 32 + 16 + nn] = acc1[r] + sBep[16 + nn] + e1;
        }
    }
    __syncthreads();

    // phase 3: out = gnf @ W_nodes + b_nodes   (N = 64 -> 4 tiles)
    if (wave < 2) {
        v8f acc[4];
        #pragma unroll
        for (int t = 0; t < 4; ++t)
            for (int r = 0; r < 8; ++r) acc[t][r] = sBn[t * 16 + nn];
        #pragma unroll
        for (int kk = 0; kk < 8; ++kk) {
            int k0 = kk * 4 + 2 * h;
            v2f A; A.x = sGnf[(m0 + nn) * 32 + k0];
                   A.y = sGnf[(m0 + nn) * 32 + k0 + 1];
            #pragma unroll
            for (int t = 0; t < 4; ++t) {
                v2f B; B.x = sWn[k0 * 64 + t * 16 + nn];
                       B.y = sWn[(k0 + 1) * 64 + t * 16 + nn];
                acc[t] = wmma_f32(A, B, acc[t]);
            }
        }
        #pragma unroll
        for (int t = 0; t < 4; ++t)
            for (int r = 0; r < 8; ++r) {
                int row = m0 + 8 * h + r;
                if (row < NPG_)
                    __builtin_nontemporal_store(acc[t][r],
                        &out_node[(g * NPG_ + row) * DH_ + t * 16 + nn]);
            }
    }
}

// ---------------------------------------------------------------------------
// Edge kernel: 8 waves/block, TPW_ x 16 edges/wave (B fragments, bias and
// gaussian params hoisted to registers, amortized over TPW_ tiles).
// Edge reversal per reference: row i = dst local, col j = src local,
// edge_type = z[dst]*10 + z[src].
// ---------------------------------------------------------------------------
__global__ __launch_bounds__(256) void edge_kernel(
    const int* __restrict__ z, const float* __restrict__ pos,
    const int* __restrict__ edge_index,
    const float* __restrict__ means, const float* __restrict__ stds,
    const float* __restrict__ gmul, const float* __restrict__ gbias,
    const float* __restrict__ W_e, const float* __restrict__ b_e,
    float* __restrict__ out_edge)
{
    __shared__ float sW[32 * 64];
    __shared__ float sB[64];
    __shared__ float sMean[K_], sQ[K_], sInvNorm[K_];

    const int tid = threadIdx.x;
    for (int p = tid; p < 32 * 64; p += 256) sW[p] = W_e[p];
    if (tid < 64) sB[tid] = b_e[tid];
    if (tid < K_) {
        float s  = fabsf(stds[tid]) + 1e-5f;
        float is = 1.0f / s;
        sMean[tid]    = means[tid];
        sQ[tid]       = NEG_HALF_LOG2E * is * is;
        sInvNorm[tid] = INV_SQRT_2PI * is;
    }
    __syncthreads();

    const int wave = tid >> 5, lane = tid & 31;
    const int h = lane >> 4, nn = lane & 15;

    // hoist B fragments: this lane supplies K rows {4kk+2h, 4kk+2h+1}
    v2f Breg[8][4];
    #pragma unroll
    for (int kk = 0; kk < 8; ++kk) {
        int k0 = kk * 4 + 2 * h;
        #pragma unroll
        for (int t = 0; t < 4; ++t) {
            Breg[kk][t].x = sW[k0 * 64 + t * 16 + nn];
            Breg[kk][t].y = sW[(k0 + 1) * 64 + t * 16 + nn];
        }
    }
    float biasn[4];
    #pragma unroll
    for (int t = 0; t < 4; ++t) biasn[t] = sB[t * 16 + nn];

    // per-lane gaussian params for its 16 owned k values
    float mn[16], qv[16], nv[16];
    #pragma unroll
    for (int kk = 0; kk < 8; ++kk) {
        int k0 = kk * 4 + 2 * h;
        mn[2 * kk] = sMean[k0];     mn[2 * kk + 1] = sMean[k0 + 1];
        qv[2 * kk] = sQ[k0];        qv[2 * kk + 1] = sQ[k0 + 1];
        nv[2 * kk] = sInvNorm[k0];  nv[2 * kk + 1] = sInvNorm[k0 + 1];
    }

    const int base = (blockIdx.x * 8 + wave) * (TPW_ * 16);

    for (int tile = 0; tile < TPW_; ++tile) {
        const int e0 = base + tile * 16;
        const int e  = e0 + nn;               // A layout: lane&15 = M row (edge)

        int src = edge_index[e];              // edge_index[0][e]
        int dst = edge_index[E_ + e];         // edge_index[1][e]
        float dx = pos[src * 3 + 0] - pos[dst * 3 + 0];
        float dy = pos[src * 3 + 1] - pos[dst * 3 + 1];
        float dz = pos[src * 3 + 2] - pos[dst * 3 + 2];
        float sq = dx * dx + dy * dy + dz * dz;
        float dist = (sq == 0.0f) ? 0.0f : sqrtf(sq);
        int et = z[dst] * 10 + z[src];
        float x = gmul[et] * dist + gbias[et];

        float feat[16];
        #pragma unroll
        for (int u = 0; u < 16; ++u) {
            float d = x - mn[u];
            feat[u] = exp2f(qv[u] * d * d) * nv[u];   // v_exp_f32
        }

        v8f acc[4];
        #pragma unroll
        for (int t = 0; t < 4; ++t)
            for (int r = 0; r < 8; ++r) acc[t][r] = biasn[t];

        #pragma unroll
        for (int kk = 0; kk < 8; ++kk) {
            v2f A; A.x = feat[2 * kk]; A.y = feat[2 * kk + 1];
            #pragma unroll
            for (int t = 0; t < 4; ++t)
                acc[t] = wmma_f32(A, Breg[kk][t], acc[t]);
        }

        #pragma unroll
        for (int t = 0; t < 4; ++t)
            for (int r = 0; r < 8; ++r)
                __builtin_nontemporal_store(acc[t][r],
                    &out_edge[(e0 + 8 * h + r) * DH_ + t * 16 + nn]);
    }
}

extern "C" void kernel_launch(void* const* d_in, const int* in_sizes, int n_in,
                              void* d_out, int out_size, void* d_ws, size_t ws_size,
                              hipStream_t stream) {
    (void)in_sizes; (void)n_in; (void)out_size; (void)d_ws; (void)ws_size;
    const int*   z          = (const int*)d_in[0];
    const float* pos        = (const float*)d_in[1];
    // d_in[2] = batch_mapping (implicit: i / NPG_), unused
    const int*   edge_index = (const int*)d_in[3];
    const float* atom_emb   = (const float*)d_in[4];
    const float* W_ep       = (const float*)d_in[5];
    const float* b_ep       = (const float*)d_in[6];
    const float* gbf_means  = (const float*)d_in[7];
    const float* gbf_stds   = (const float*)d_in[8];
    const float* gbf_mul    = (const float*)d_in[9];
    const float* gbf_bias   = (const float*)d_in[10];
    const float* W_nodes    = (const float*)d_in[11];
    const float* b_nodes    = (const float*)d_in[12];
    const float* W_edges    = (const float*)d_in[13];
    const float* b_edges    = (const float*)d_in[14];

    float* out_node = (float*)d_out;                     // N_ * 64
    float* out_edge = (float*)d_out + (size_t)N_ * DH_;  // E_ * 64

    node_kernel<<<G_, 256, 0, stream>>>(
        z, pos, atom_emb, W_ep, b_ep, gbf_means, gbf_stds, gbf_mul, gbf_bias,
        W_nodes, b_nodes, out_node);

    edge_kernel<<<E_ / (8 * TPW_ * 16), 256, 0, stream>>>(
        z, pos, edge_index, gbf_means, gbf_stds, gbf_mul, gbf_bias,
        W_edges, b_edges, out_edge);
}